// QKNet_54219667145470
// MI455X (gfx1250) — compile-verified
//
#include <hip/hip_runtime.h>
#include <hip/hip_bf16.h>

typedef __attribute__((ext_vector_type(2))) float f32x2;
typedef __attribute__((ext_vector_type(8))) float f32x8;
typedef __attribute__((ext_vector_type(4))) unsigned int u32x4;
typedef __attribute__((ext_vector_type(8))) int i32x8;

// ---------------------------------------------------------------------------
// WMMA fp32 16x16x4 (CDNA5 V_WMMA_F32_16X16X4_F32) — codegen confirmed R1.
// A: 16x4 fp32, lane L holds row (L&15), K elems 2*(L>>4), 2*(L>>4)+1.
// B: 4x16 fp32, lane L holds col (L&15), same K pair.
// C/D: 8 VGPRs, element r -> (m = r + 8*(L>>4), n = L&15).
// ---------------------------------------------------------------------------
__device__ __forceinline__ f32x8 wmma_f32_16x16x4(f32x2 a, f32x2 b, f32x8 c) {
    return __builtin_amdgcn_wmma_f32_16x16x4_f32(
        false, a, false, b, (short)0, c, false, false);
}

// ---------------------------------------------------------------------------
// Tensor Data Mover: DMA a 2D fp32 tile (rows x cols, cols==32) from global
// memory into LDS, with LDS row padding of 4 DWORDs after every 32 DWORDs
// (-> padded row stride 36 floats = 144 B, 16B-aligned, conflict-free).
// Descriptor per CDNA5 ISA §8.3 (group0) / §8.4 (group1). Not in a cluster,
// so workgroup_mask = 0. Tracked by TENSORcnt.
// ---------------------------------------------------------------------------
__device__ __forceinline__ unsigned lds_off(const void* p) {
    // generic LDS pointer = {SHARED_BASE aperture, lds_byte_offset}; low 32
    // bits are the LDS byte address (ISA §10.2 aperture rules).
    return (unsigned)(unsigned long long)p;
}

__device__ __forceinline__ void tdm_load_tile_f32(
    unsigned lds_byte, const float* gsrc,
    unsigned rows, unsigned cols, unsigned long long row_stride_elems)
{
    unsigned long long ga = (unsigned long long)gsrc;
    u32x4 g0;
    g0[0] = 1u;                                         // count=1, user mode
    g0[1] = lds_byte;                                   // lds_addr
    g0[2] = (unsigned)ga;                               // global_addr[31:0]
    g0[3] = (unsigned)((ga >> 32) & 0x01FFFFFFull)      // global_addr[56:32]
          | (2u << 30);                                 // type = 2 ("image")
    i32x8 g1;
    g1[0] = (2 << 16)                                   // data_size = 4 B
          | (1 << 20)                                   // pad_enable
          | (4 << 22)                                   // pad_interval: 32 DW
          | (3 << 25);                                  // pad_amount: 4 DW
    g1[1] = (int)((cols & 0xFFFFu) << 16);              // tensor_dim0[15:0]
    g1[2] = (int)(((cols >> 16) & 0xFFFFu)              // tensor_dim0[31:16]
          | ((rows & 0xFFFFu) << 16));                  // tensor_dim1[15:0]
    g1[3] = (int)(((rows >> 16) & 0xFFFFu)              // tensor_dim1[31:16]
          | ((cols & 0xFFFFu) << 16));                  // tile_dim0 = cols
    g1[4] = (int)(rows & 0xFFFFu);                      // tile_dim1, tile_dim2=0
    g1[5] = (int)(row_stride_elems & 0xFFFFFFFFull);    // dim0_stride[31:0]
    g1[6] = (int)((row_stride_elems >> 32) & 0xFFFFull);// dim0_stride[47:32]
    g1[7] = 0;
    asm volatile("tensor_load_to_lds %0, %1" :: "s"(g0), "s"(g1) : "memory");
}

template <int N>
__device__ __forceinline__ void tdm_wait() {
#if defined(__has_builtin)
#if __has_builtin(__builtin_amdgcn_s_wait_tensorcnt)
    __builtin_amdgcn_s_wait_tensorcnt((short)N);
#else
    asm volatile("s_wait_tensorcnt %0" :: "i"(N) : "memory");
#endif
#else
    asm volatile("s_wait_tensorcnt %0" :: "i"(N) : "memory");
#endif
}

// ===========================================================================
// conv1 (3->32, 5x5, pad 2) + bias + relu + 2x2 maxpool, fused.
// grid: (7, 7, B*32), block (16,16). pooled output 112x112.
// ===========================================================================
__global__ __launch_bounds__(256) void conv1_relu_pool(
    const float* __restrict__ x, const float* __restrict__ w,
    const float* __restrict__ bias, float* __restrict__ out)
{
    const int b  = blockIdx.z >> 5;
    const int co = blockIdx.z & 31;
    const int py0 = blockIdx.y * 16, px0 = blockIdx.x * 16;
    const int tid = threadIdx.y * 16 + threadIdx.x;

    __shared__ float sIn[3][36][36];   // conv region 32x32 + halo 4
    __shared__ float sW[3][5][5];

    if (tid < 75) ((float*)sW)[tid] = w[co * 75 + tid];

    const int gy0 = 2 * py0 - 2, gx0 = 2 * px0 - 2;
    for (int i = tid; i < 3 * 36 * 36; i += 256) {
        int ci = i / 1296, rem = i % 1296;
        int r = rem / 36, cc = rem % 36;
        int gy = gy0 + r, gx = gx0 + cc;
        float v = 0.0f;
        if (gy >= 0 && gy < 224 && gx >= 0 && gx < 224)
            v = x[((size_t)(b * 3 + ci) * 224 + gy) * 224 + gx];
        sIn[ci][r][cc] = v;
    }
    __syncthreads();

    const float bsv = bias[co];
    float m = -3.4e38f;
    #pragma unroll
    for (int dy = 0; dy < 2; ++dy) {
        #pragma unroll
        for (int dx = 0; dx < 2; ++dx) {
            const int r0 = 2 * threadIdx.y + dy;
            const int c0 = 2 * threadIdx.x + dx;
            float acc = bsv;
            #pragma unroll
            for (int ci = 0; ci < 3; ++ci)
                #pragma unroll
                for (int ky = 0; ky < 5; ++ky)
                    #pragma unroll
                    for (int kx = 0; kx < 5; ++kx)
                        acc += sIn[ci][r0 + ky][c0 + kx] * sW[ci][ky][kx];
            acc = fmaxf(acc, 0.0f);
            m = fmaxf(m, acc);
        }
    }
    out[((size_t)(b * 32 + co) * 112 + py0 + threadIdx.y) * 112 + px0 + threadIdx.x] = m;
}

// ===========================================================================
// conv2 (32->64, 5x5, pad 2 on 112x112) + bias + relu + 2x2 maxpool, fused.
// grid: (4, 4, B*64), block (16,16). pooled output 56x56 (edge tiles masked).
// ===========================================================================
__global__ __launch_bounds__(256) void conv2_relu_pool(
    const float* __restrict__ x, const float* __restrict__ w,
    const float* __restrict__ bias, float* __restrict__ out)
{
    const int b  = blockIdx.z >> 6;
    const int co = blockIdx.z & 63;
    const int py0 = blockIdx.y * 16, px0 = blockIdx.x * 16;
    const int tid = threadIdx.y * 16 + threadIdx.x;
    const int py = py0 + threadIdx.y, px = px0 + threadIdx.x;

    __shared__ float sIn[4][36][36];
    __shared__ float sW[4][25];

    float acc[4];
    const float bsv = bias[co];
    #pragma unroll
    for (int p = 0; p < 4; ++p) acc[p] = bsv;

    const int gy0 = 2 * py0 - 2, gx0 = 2 * px0 - 2;

    for (int cc0 = 0; cc0 < 32; cc0 += 4) {
        __syncthreads();   // protect previous iteration's LDS reads
        for (int i = tid; i < 4 * 36 * 36; i += 256) {
            int ci = i / 1296, rem = i % 1296;
            int r = rem / 36, cc = rem % 36;
            int gy = gy0 + r, gx = gx0 + cc;
            float v = 0.0f;
            if (gy >= 0 && gy < 112 && gx >= 0 && gx < 112)
                v = x[((size_t)(b * 32 + cc0 + ci) * 112 + gy) * 112 + gx];
            sIn[ci][r][cc] = v;
        }
        if (tid < 100)
            sW[tid / 25][tid % 25] = w[(size_t)(co * 32 + cc0 + tid / 25) * 25 + tid % 25];
        __syncthreads();

        #pragma unroll
        for (int dy = 0; dy < 2; ++dy)
            #pragma unroll
            for (int dx = 0; dx < 2; ++dx) {
                const int r0 = 2 * threadIdx.y + dy;
                const int c0 = 2 * threadIdx.x + dx;
                float a = 0.0f;
                #pragma unroll
                for (int ci = 0; ci < 4; ++ci)
                    #pragma unroll
                    for (int ky = 0; ky < 5; ++ky)
                        #pragma unroll
                        for (int kx = 0; kx < 5; ++kx)
                            a += sIn[ci][r0 + ky][c0 + kx] * sW[ci][ky * 5 + kx];
                acc[dy * 2 + dx] += a;
            }
    }

    if (py < 56 && px < 56) {
        float m = -3.4e38f;
        #pragma unroll
        for (int p = 0; p < 4; ++p) m = fmaxf(m, fmaxf(acc[p], 0.0f));
        out[((size_t)(b * 64 + co) * 56 + py) * 56 + px] = m;
    }
}

// ===========================================================================
// knn similarity + argmax. One block per channel c.
// S[b,k] = sum_d x[b,c,d] * center[c,k,d];  idx[b,c] = argmax_k S[b,k]
// (1/||x|| is a positive per-(b,c) constant -> argmax invariant)
// M=32(b), N=64(k) via v_wmma_f32_16x16x4_f32; TDM double-buffered LDS tiles.
// ===========================================================================
__global__ __launch_bounds__(256) void knn_sim_argmax(
    const float* __restrict__ x, const float* __restrict__ center,
    int C, int D, int* __restrict__ idx_out)
{
    const int BK = 32, BKP = 36;        // TDM-padded LDS row (144 B)
    const int c = blockIdx.x;
    const int tid = threadIdx.x;
    const int lane = tid & 31, wv = tid >> 5;
    const int half = lane >> 4, l16 = lane & 15;
    const int m0 = (wv >> 2) * 16;      // 0 or 16 (batch rows)
    const int n0 = (wv & 3) * 16;       // 0,16,32,48 (centers)

    __shared__ float As[2][32 * 36];
    __shared__ float Bs[2][64 * 36];
    __shared__ float Ss[32][65];

    const float* __restrict__ A0 = x + (size_t)c * D;           // row stride C*D
    const float* __restrict__ B0 = center + (size_t)c * 64 * D; // row stride D
    const unsigned long long strA = (unsigned long long)C * (unsigned long long)D;
    const int nch = D / BK;             // D is a multiple of 32

    if (wv == 0) {
        tdm_load_tile_f32(lds_off(&As[0][0]), A0, 32, BK, strA);
        tdm_load_tile_f32(lds_off(&Bs[0][0]), B0, 64, BK, (unsigned long long)D);
    }

    f32x8 acc = {};
    for (int ch = 0; ch < nch; ++ch) {
        __syncthreads();                 // everyone done reading buf[(ch+1)&1]
        if (wv == 0) {
            if (ch + 1 < nch) {
                const int d0 = (ch + 1) * BK;
                tdm_load_tile_f32(lds_off(&As[(ch + 1) & 1][0]), A0 + d0, 32, BK, strA);
                tdm_load_tile_f32(lds_off(&Bs[(ch + 1) & 1][0]), B0 + d0, 64, BK,
                                  (unsigned long long)D);
                tdm_wait<2>();           // 2 newest outstanding -> chunk ch resident
            } else {
                tdm_wait<0>();
            }
        }
        __syncthreads();                 // buf[ch&1] visible to all waves
        const float* __restrict__ Ab = As[ch & 1];
        const float* __restrict__ Bb = Bs[ch & 1];
        #pragma unroll
        for (int k = 0; k < BK; k += 4) {
            f32x2 a  = *(const f32x2*)&Ab[(m0 + l16) * BKP + k + 2 * half];
            f32x2 bv = *(const f32x2*)&Bb[(n0 + l16) * BKP + k + 2 * half];
            acc = wmma_f32_16x16x4(a, bv, acc);
        }
    }

    __syncthreads();
    #pragma unroll
    for (int r = 0; r < 8; ++r)
        Ss[m0 + r + 8 * half][n0 + l16] = acc[r];
    __syncthreads();

    if (tid < 32) {   // argmin(1-sim) == first-occurrence argmax(sim)
        float best = Ss[tid][0];
        int bi = 0;
        for (int k = 1; k < 64; ++k) {
            float v = Ss[tid][k];
            if (v > best) { best = v; bi = k; }
        }
        idx_out[tid * C + c] = bi;
    }
}

// q[b,c,:] = center[c, idx[b,c], :]
__global__ __launch_bounds__(256) void gather_centers(
    const float* __restrict__ center, const int* __restrict__ idx,
    int C, int D, float* __restrict__ q)
{
    const int bc = blockIdx.x;          // b*C + c
    const int c = bc % C;
    const int id = idx[bc];
    const float* __restrict__ src = center + ((size_t)c * 64 + id) * D;
    float* __restrict__ dst = q + (size_t)bc * D;
    for (int d = threadIdx.x; d < D; d += blockDim.x) dst[d] = src[d];
}

// ===========================================================================
// fc1: S[32,1024] += A[32,200704] * W[1024,200704]^T
// split-K (49 x 4096) + fp32 atomics; TDM double-buffered tiles; 8 waves.
// fc1_w is 822 MB fp32 -> HBM-bound (~35 us floor at 23.3 TB/s); TDM streams
// it into LDS while waves stay in the WMMA loop.
// ===========================================================================
__global__ __launch_bounds__(256) void fc1_gemm(
    const float* __restrict__ A, const float* __restrict__ W,
    float* __restrict__ accbuf)
{
    const int Kd = 200704, N = 1024, BK = 32, BKP = 36, KCH = 4096;
    const int n0g = blockIdx.x * 64;
    const size_t k0g = (size_t)blockIdx.y * KCH;
    const int tid = threadIdx.x;
    const int lane = tid & 31, wv = tid >> 5;
    const int half = lane >> 4, l16 = lane & 15;
    const int m0 = (wv >> 2) * 16;
    const int n0 = (wv & 3) * 16;

    __shared__ float As[2][32 * 36];
    __shared__ float Bs[2][64 * 36];

    const float* __restrict__ A0 = A + k0g;                    // row stride Kd
    const float* __restrict__ B0 = W + (size_t)n0g * Kd + k0g; // row stride Kd
    const int nch = KCH / BK;   // 128

    if (wv == 0) {
        tdm_load_tile_f32(lds_off(&As[0][0]), A0, 32, BK, (unsigned long long)Kd);
        tdm_load_tile_f32(lds_off(&Bs[0][0]), B0, 64, BK, (unsigned long long)Kd);
    }

    f32x8 acc = {};
    for (int ch = 0; ch < nch; ++ch) {
        __syncthreads();
        if (wv == 0) {
            if (ch + 1 < nch) {
                const int d0 = (ch + 1) * BK;
                tdm_load_tile_f32(lds_off(&As[(ch + 1) & 1][0]), A0 + d0, 32, BK,
                                  (unsigned long long)Kd);
                tdm_load_tile_f32(lds_off(&Bs[(ch + 1) & 1][0]), B0 + d0, 64, BK,
                                  (unsigned long long)Kd);
                tdm_wait<2>();
            } else {
                tdm_wait<0>();
            }
        }
        __syncthreads();
        const float* __restrict__ Ab = As[ch & 1];
        const float* __restrict__ Bb = Bs[ch & 1];
        #pragma unroll
        for (int k = 0; k < BK; k += 4) {
            f32x2 a  = *(const f32x2*)&Ab[(m0 + l16) * BKP + k + 2 * half];
            f32x2 bv = *(const f32x2*)&Bb[(n0 + l16) * BKP + k + 2 * half];
            acc = wmma_f32_16x16x4(a, bv, acc);
        }
    }
    #pragma unroll
    for (int r = 0; r < 8; ++r)
        atomicAdd(&accbuf[(size_t)(m0 + r + 8 * half) * N + n0g + n0 + l16], acc[r]);
}

__global__ void zero_f32(float* __restrict__ p, int n) {
    int i = blockIdx.x * blockDim.x + threadIdx.x;
    if (i < n) p[i] = 0.0f;
}

__global__ void fc1_bias_relu(const float* __restrict__ acc,
                              const float* __restrict__ bias,
                              float* __restrict__ out)
{
    int i = blockIdx.x * blockDim.x + threadIdx.x;
    if (i < 32 * 1024) {
        int n = i & 1023;
        out[i] = fmaxf(acc[i] + bias[n], 0.0f);
    }
}

// fc2: out[32,10] = h[32,1024] * w[10,1024]^T + b
__global__ void fc2_kernel(const float* __restrict__ h, const float* __restrict__ w,
                           const float* __restrict__ bias, float* __restrict__ out)
{
    int t = blockIdx.x * blockDim.x + threadIdx.x;
    if (t >= 320) return;
    int b = t / 10, j = t % 10;
    float s = bias[j];
    const float* __restrict__ hr = h + (size_t)b * 1024;
    const float* __restrict__ wr = w + (size_t)j * 1024;
    for (int n = 0; n < 1024; ++n) s += hr[n] * wr[n];
    out[b * 10 + j] = s;
}

// ===========================================================================
extern "C" void kernel_launch(void* const* d_in, const int* in_sizes, int n_in,
                              void* d_out, int out_size, void* d_ws, size_t ws_size,
                              hipStream_t stream) {
    const float* x       = (const float*)d_in[0];
    const float* conv1_w = (const float*)d_in[1];
    const float* conv1_b = (const float*)d_in[2];
    const float* conv2_w = (const float*)d_in[3];
    const float* conv2_b = (const float*)d_in[4];
    const float* fc1_w   = (const float*)d_in[5];
    const float* fc1_b   = (const float*)d_in[6];
    const float* fc2_w   = (const float*)d_in[7];
    const float* fc2_b   = (const float*)d_in[8];
    const float* center0 = (const float*)d_in[9];
    const float* center1 = (const float*)d_in[10];
    float* out = (float*)d_out;

    // workspace carving
    const size_t N1 = (size_t)32 * 32 * 112 * 112;   // 12,845,056 floats
    const size_t N2 = (size_t)32 * 64 * 56 * 56;     //  6,422,528 floats
    float* pool1  = (float*)d_ws;
    float* q1     = pool1 + N1;
    float* pool2  = q1 + N1;
    float* q2     = pool2 + N2;
    float* fc1acc = q2 + N2;
    float* fc1out = fc1acc + 32 * 1024;
    int*   idx0   = (int*)(fc1out + 32 * 1024);
    int*   idx1   = idx0 + 32 * 32;

    // 1) conv1 + relu + pool -> pool1 (32,32,112,112)
    conv1_relu_pool<<<dim3(7, 7, 32 * 32), dim3(16, 16), 0, stream>>>(
        x, conv1_w, conv1_b, pool1);

    // 2) knn0: similarity GEMM + argmax, then gather -> q1
    knn_sim_argmax<<<32, 256, 0, stream>>>(pool1, center0, 32, 112 * 112, idx0);
    gather_centers<<<32 * 32, 256, 0, stream>>>(center0, idx0, 32, 112 * 112, q1);

    // 3) conv2 + relu + pool -> pool2 (32,64,56,56)
    conv2_relu_pool<<<dim3(4, 4, 32 * 64), dim3(16, 16), 0, stream>>>(
        q1, conv2_w, conv2_b, pool2);

    // 4) knn1 + gather -> q2 (== flattened fc input, layout (b, c*3136+s))
    knn_sim_argmax<<<64, 256, 0, stream>>>(pool2, center1, 64, 56 * 56, idx1);
    gather_centers<<<32 * 64, 256, 0, stream>>>(center1, idx1, 64, 56 * 56, q2);

    // 5) fc1 (split-K WMMA GEMM, TDM-fed, atomic reduction) + bias + relu
    zero_f32<<<(32 * 1024 + 255) / 256, 256, 0, stream>>>(fc1acc, 32 * 1024);
    fc1_gemm<<<dim3(16, 49), 256, 0, stream>>>(q2, fc1_w, fc1acc);
    fc1_bias_relu<<<(32 * 1024 + 255) / 256, 256, 0, stream>>>(fc1acc, fc1_b, fc1out);

    // 6) fc2 -> out (32,10)
    fc2_kernel<<<1, 320, 0, stream>>>(fc1out, fc2_w, fc2_b, out);
}